// base_projection_82463372083832
// MI455X (gfx1250) — compile-verified
//
#include <hip/hip_runtime.h>

// CDNA5 / gfx1250: Lie-algebra projection + complex 16x16 matrix exponential.
//   AX[n,c] = sum_i dX[n,i] * (A[i,c] - A[i,c]^H)        (anti-Hermitian)
//   out[n,c] = expm(AX[n,c])  via scaling(2^-9) + order-8 Taylor-Horner + 9 squarings
// One wave32 per (n,c) matrix; all 16x16 complex matmuls done with
// V_WMMA_F32_16X16X4_F32 (fp32 WMMA is required: 9 squarings amplify error ~512x,
// f16 operands would be unacceptable). Complex product = 4 real WMMA chains.

typedef float v2f __attribute__((ext_vector_type(2)));
typedef float v8f __attribute__((ext_vector_type(8)));

#define N_PATHS   8192
#define IN_SIZE   64
#define CHANNELS  8
#define SCALE_S   9      // squarings: 2^9 = 512 > ||AX|| ~ 75
#define TAYLOR_K  8      // Taylor order after scaling (||S|| ~ 0.15)
#define CHUNK_I   16     // input-dim slices staged in LDS per pass

static __device__ __forceinline__ v8f wmma4(v2f a, v2f b, v8f c) {
  // D(16x16,f32) = A(16x4,f32) * B(4x16,f32) + C
  return __builtin_amdgcn_wmma_f32_16x16x4_f32(
      /*neg_a=*/false, a, /*neg_b=*/false, b,
      /*c_mod=*/(short)0, c, /*reuse_a=*/false, /*reuse_b=*/false);
}

static __device__ __forceinline__ float lane_bcast(float x, int lane) {
  return __uint_as_float(__builtin_amdgcn_readlane(__float_as_uint(x), lane));
}

// C = (Ar + i*Ai)(Br + i*Bi): Cr = Ar*Br + (-Ai)*Bi ; Ci = Ar*Bi + Ai*Br
// Each term is a 16x16x16 matmul = 4 chained 16x16x4 WMMAs (K slices).
static __device__ __forceinline__ void cmul16(
    const v2f* Ar, const v2f* Ai, const v2f* nAi,
    const v2f* Br, const v2f* Bi, v8f& Cr, v8f& Ci)
{
  v8f cr = {}; v8f ci = {};
#pragma unroll
  for (int j = 0; j < 4; ++j) cr = wmma4(Ar[j], Br[j], cr);
#pragma unroll
  for (int j = 0; j < 4; ++j) cr = wmma4(nAi[j], Bi[j], cr);
#pragma unroll
  for (int j = 0; j < 4; ++j) ci = wmma4(Ar[j], Bi[j], ci);
#pragma unroll
  for (int j = 0; j < 4; ++j) ci = wmma4(Ai[j], Br[j], ci);
  Cr = cr; Ci = ci;
}

__global__ void __launch_bounds__(256)
lie_expm_kernel(const float* __restrict__ dX,
                const float2* __restrict__ A,
                float2* __restrict__ out)
{
  // 32 KB staging for CHUNK_I input-slices of Ah[:,c] + 2 KB/wave matmul scratch
  __shared__ __align__(16) float2 sAh[CHUNK_I * 256];
  __shared__ __align__(16) float2 sMat[8 * 256];

  const int tid  = threadIdx.x;
  const int wave = tid >> 5;
  const int lane = tid & 31;
  const int half = lane >> 4;   // C/D layout: lanes 16-31 hold M+8 rows
  const int l16  = lane & 15;   // N column (C/D, B) or M row (A operand)
  const int c    = blockIdx.y;
  const int b    = blockIdx.x * 8 + wave;

  float2* wbuf = &sMat[wave * 256];   // wave-private 16x16 complex tile

  // per-lane slice of dX[b,:]; broadcast later via v_readlane
  const float dx0 = dX[b * IN_SIZE + lane];
  const float dx1 = dX[b * IN_SIZE + 32 + lane];

  // diagonal indicator for "+= I" in C/D layout: element (v+8*half, l16)
  float dI[8];
#pragma unroll
  for (int v = 0; v < 8; ++v) dI[v] = ((v + 8 * half) == l16) ? 1.0f : 0.0f;

  // ---------- einsum: AX (C-layout accumulators, complex) ----------
  v8f AXr = {}; v8f AXi = {};
  for (int chunk = 0; chunk < IN_SIZE / CHUNK_I; ++chunk) {
    // cooperative staging, anti-Hermitian projection fused:
    // Ah[m,k] = (Ar[m,k]-Ar[k,m]) + i*(Ai[m,k]+Ai[k,m])
    for (int e = tid; e < CHUNK_I * 256; e += 256) {
      const int il = e >> 8, rem = e & 255, m = rem >> 4, k = rem & 15;
      const int i = chunk * CHUNK_I + il;
      const size_t base = ((size_t)i * CHANNELS + c) * 256;
      const float2 amk = A[base + m * 16 + k];
      const float2 akm = A[base + k * 16 + m];
      sAh[e] = make_float2(amk.x - akm.x, amk.y + akm.y);
    }
    __syncthreads();
#pragma unroll
    for (int il = 0; il < CHUNK_I; ++il) {
      const int i = chunk * CHUNK_I + il;
      const float d = lane_bcast((i < 32) ? dx0 : dx1, i & 31);
#pragma unroll
      for (int v = 0; v < 8; ++v) {
        const float2 ah = sAh[il * 256 + (v + 8 * half) * 16 + l16];
        AXr[v] = fmaf(d, ah.x, AXr[v]);
        AXi[v] = fmaf(d, ah.y, AXi[v]);
      }
    }
    __syncthreads();
  }

  // ---------- S = AX * 2^-s ----------
  const float sc = 1.0f / (float)(1 << SCALE_S);
  v8f Sr, Si;
#pragma unroll
  for (int v = 0; v < 8; ++v) { Sr[v] = AXr[v] * sc; Si[v] = AXi[v] * sc; }

  // helpers: C-layout <-> LDS row-major tile, A/B operand slice loads.
  auto storeC = [&](const v8f& r, const v8f& im) {
#pragma unroll
    for (int v = 0; v < 8; ++v)
      wbuf[(v + 8 * half) * 16 + l16] = make_float2(r[v], im[v]);
  };
  // A operand (16x4 slices): lane -> row M=l16; VGPR0/1 -> K = 4j+2*half (+1)
  auto loadA = [&](v2f* Ar, v2f* Ai, v2f* nAi) {
#pragma unroll
    for (int j = 0; j < 4; ++j) {
      const float4 q = *reinterpret_cast<const float4*>(
          &wbuf[l16 * 16 + 4 * j + 2 * half]);        // 2 adjacent complex
      v2f ar = {q.x, q.z}; v2f ai = {q.y, q.w}; v2f na = {-q.y, -q.w};
      Ar[j] = ar; Ai[j] = ai; nAi[j] = na;
    }
  };
  // B operand (4x16 slices): lane -> col N=l16; VGPR0/1 -> K = 4j+2*half (+1)
  auto loadB = [&](v2f* Br, v2f* Bi) {
#pragma unroll
    for (int j = 0; j < 4; ++j) {
      const int k0 = 4 * j + 2 * half;
      const float2 e0 = wbuf[k0 * 16 + l16];
      const float2 e1 = wbuf[(k0 + 1) * 16 + l16];
      v2f br = {e0.x, e1.x}; v2f bi = {e0.y, e1.y};
      Br[j] = br; Bi[j] = bi;
    }
  };

  // S in A-operand registers, fixed across the Horner recurrence
  storeC(Sr, Si);
  v2f SAr[4], SAi[4], SnAi[4];
  loadA(SAr, SAi, SnAi);

  // T = I + S/K
  v8f Tr, Ti;
  const float invK = 1.0f / (float)TAYLOR_K;
#pragma unroll
  for (int v = 0; v < 8; ++v) { Tr[v] = Sr[v] * invK + dI[v]; Ti[v] = Si[v] * invK; }

  // Horner: T = I + (S*T)/k , k = K-1 .. 1
  for (int k = TAYLOR_K - 1; k >= 1; --k) {
    storeC(Tr, Ti);
    v2f Br[4], Bi[4];
    loadB(Br, Bi);
    v8f Cr, Ci;
    cmul16(SAr, SAi, SnAi, Br, Bi, Cr, Ci);
    const float ik = 1.0f / (float)k;
#pragma unroll
    for (int v = 0; v < 8; ++v) { Tr[v] = Cr[v] * ik + dI[v]; Ti[v] = Ci[v] * ik; }
  }

  // Squaring: E = E*E, s times
  for (int s = 0; s < SCALE_S; ++s) {
    storeC(Tr, Ti);
    v2f Ar[4], Ai[4], nAi[4];
    loadA(Ar, Ai, nAi);
    v2f Br[4], Bi[4];
    loadB(Br, Bi);
    v8f Cr, Ci;
    cmul16(Ar, Ai, nAi, Br, Bi, Cr, Ci);
#pragma unroll
    for (int v = 0; v < 8; ++v) { Tr[v] = Cr[v]; Ti[v] = Ci[v]; }
  }

  // ---------- write result (complex64, row-major 16x16 per (b,c)) ----------
  const size_t obase = ((size_t)b * CHANNELS + c) * 256;
#pragma unroll
  for (int v = 0; v < 8; ++v)
    out[obase + (v + 8 * half) * 16 + l16] = make_float2(Tr[v], Ti[v]);
}

extern "C" void kernel_launch(void* const* d_in, const int* in_sizes, int n_in,
                              void* d_out, int out_size, void* d_ws, size_t ws_size,
                              hipStream_t stream) {
  (void)in_sizes; (void)n_in; (void)out_size; (void)d_ws; (void)ws_size;
  const float*  dX = (const float*)d_in[0];
  const float2* A  = (const float2*)d_in[1];   // complex64 -> interleaved float2
  float2* out = (float2*)d_out;                // complex64 output

  dim3 grid(N_PATHS / 8, CHANNELS);            // 8 waves/block, 1 matrix/wave
  lie_expm_kernel<<<grid, 256, 0, stream>>>(dX, A, out);
}